// RelationalGraphAttention_63943473103184
// MI455X (gfx1250) — compile-verified
//
#include <hip/hip_runtime.h>
#include <hip/hip_bf16.h>
#include <math.h>

typedef __attribute__((ext_vector_type(2))) float v2f;
typedef __attribute__((ext_vector_type(8))) float v8f;

#define M_NODES 16384
#define N_NBR   32
#define D_DIM   128
#define R_REL   237
#define ATT_SCALE 0.25f     /* Dh^-0.5 = 16^-0.5 */
#define LN_EPS  1e-5f

// ---------------------------------------------------------------------------
// GEMM: Y[M,128] = X[M,128] @ W[128,128] (+ bias). One wave per 16-row strip,
// 8 accumulator tiles (full 128-wide output row), V_WMMA_F32_16X16X4_F32.
// f32 A 16x4 layout: lane l holds row l%16; VGPR0 = K = (l/16)*2, VGPR1 = K+1.
// f32 B 4x16 layout mirrored: lane l holds col l%16; VGPR0 = K=(l/16)*2.
// C/D: VGPR i -> row i (lanes 0-15) / i+8 (lanes 16-31), col = lane%16.
// ---------------------------------------------------------------------------
__global__ __launch_bounds__(256) void rga_gemm128_kernel(
    const float* __restrict__ X, const float* __restrict__ W,
    const float* __restrict__ bias, float* __restrict__ Y, int Mrows)
{
    const int lane = threadIdx.x & 31;
    const int wave = threadIdx.x >> 5;
    const int rowTile = blockIdx.x * 8 + wave;      // 16-row tile
    const int row0 = rowTile * 16;
    if (row0 >= Mrows) return;                      // uniform per wave

    const int r  = lane & 15;                       // A row / B col within tile
    const int kg = (lane >> 4) * 2;                 // K sub-offset: 0 or 2

    int arow = row0 + r;
    if (arow >= Mrows) arow = Mrows - 1;            // clamp tail reads (R=237)

    v8f acc[8];
    #pragma unroll
    for (int t = 0; t < 8; ++t)
        #pragma unroll
        for (int i = 0; i < 8; ++i) acc[t][i] = 0.0f;

    const float* aptr = X + (size_t)arow * 128 + kg;
    for (int k0 = 0; k0 < 128; k0 += 4) {
        v2f a = *(const v2f*)(aptr + k0);
        const float* b0 = W + (size_t)(k0 + kg) * 128 + r;
        #pragma unroll
        for (int t = 0; t < 8; ++t) {
            v2f b;
            b.x = b0[t * 16];
            b.y = b0[128 + t * 16];
            acc[t] = __builtin_amdgcn_wmma_f32_16x16x4_f32(
                false, a, false, b, (short)0, acc[t], false, false);
        }
    }

    const int rbase = (lane >> 4) * 8;
    #pragma unroll
    for (int t = 0; t < 8; ++t) {
        const int col = t * 16 + r;
        const float bv = bias ? bias[col] : 0.0f;
        #pragma unroll
        for (int i = 0; i < 8; ++i) {
            const int orow = row0 + rbase + i;
            if (orow < Mrows) Y[(size_t)orow * 128 + col] = acc[t][i] + bv;
        }
    }
}

// ---------------------------------------------------------------------------
// Attention: one wave per node. Phase 1: lane = neighbor, score + softmax
// (wave32 shfl butterflies). Phase 2: lane = output dim chunk, coalesced
// V-row gathers weighted by probs from LDS.
// ---------------------------------------------------------------------------
__global__ __launch_bounds__(256) void rga_attn_kernel(
    const float* __restrict__ Q,    const float* __restrict__ Kall,
    const float* __restrict__ RelK, const float* __restrict__ Vall,
    const int* __restrict__ nbrIdx, const int* __restrict__ relIds,
    const int* __restrict__ nbrMask, float* __restrict__ attnOut)
{
    __shared__ float qs[8][128];
    __shared__ float ps[8][8 * 32];                 // [wave][h*32 + n]

    const int lane = threadIdx.x & 31;
    const int wave = threadIdx.x >> 5;
    const int m = blockIdx.x * 8 + wave;

    const float* qrow = Q + (size_t)m * 128;
    #pragma unroll
    for (int c = 0; c < 4; ++c) qs[wave][c * 32 + lane] = qrow[c * 32 + lane];
    __syncthreads();

    // ---- Phase 1: lane n scores its neighbor ----
    int nidx = nbrIdx[(size_t)m * 32 + lane];
    int nrel = relIds[(size_t)m * 32 + lane];
    const int msk = nbrMask[(size_t)m * 32 + lane];
    if (nidx < 0 || nidx >= M_NODES) nidx = 0;
    if (nrel < 0 || nrel >= R_REL)   nrel = 0;

    const float* kr = Kall + (size_t)nidx * 128;
    const float* rr = RelK + (size_t)nrel * 128;

    float sc[8];
    #pragma unroll
    for (int h = 0; h < 8; ++h) {
        const float4* k4 = (const float4*)(kr + h * 16);
        const float4* r4 = (const float4*)(rr + h * 16);
        const float4* q4 = (const float4*)(&qs[wave][h * 16]);
        float s = 0.0f;
        #pragma unroll
        for (int j = 0; j < 4; ++j) {
            float4 kk = k4[j], rv = r4[j], qq = q4[j];
            s += (kk.x + rv.x) * qq.x + (kk.y + rv.y) * qq.y
               + (kk.z + rv.z) * qq.z + (kk.w + rv.w) * qq.w;
        }
        sc[h] = s * ATT_SCALE;
    }

    #pragma unroll
    for (int h = 0; h < 8; ++h) {
        float sv = msk ? sc[h] : -3.0e38f;
        float mx = sv;
        #pragma unroll
        for (int o = 16; o > 0; o >>= 1) mx = fmaxf(mx, __shfl_xor(mx, o, 32));
        float e = msk ? __expf(sv - mx) : 0.0f;
        float sm = e;
        #pragma unroll
        for (int o = 16; o > 0; o >>= 1) sm += __shfl_xor(sm, o, 32);
        ps[wave][h * 32 + lane] = e / sm;           // >=1 valid nbr => sm > 0
    }
    __syncthreads();

    // ---- Phase 2: lane owns dims {lane, lane+32, lane+64, lane+96} ----
    const int hh = lane >> 4;                       // head = 2c + hh
    float acc0 = 0.f, acc1 = 0.f, acc2 = 0.f, acc3 = 0.f;
    for (int n = 0; n < 32; ++n) {
        const int idxn = __shfl(nidx, n, 32);
        const float* vr = Vall + (size_t)idxn * 128;
        const float p0 = ps[wave][(0 + hh) * 32 + n];
        const float p1 = ps[wave][(2 + hh) * 32 + n];
        const float p2 = ps[wave][(4 + hh) * 32 + n];
        const float p3 = ps[wave][(6 + hh) * 32 + n];
        acc0 += p0 * vr[lane];
        acc1 += p1 * vr[32 + lane];
        acc2 += p2 * vr[64 + lane];
        acc3 += p3 * vr[96 + lane];
    }
    float* orow = attnOut + (size_t)m * 128;
    orow[lane]      = acc0;
    orow[32 + lane] = acc1;
    orow[64 + lane] = acc2;
    orow[96 + lane] = acc3;
}

// ---------------------------------------------------------------------------
// Residual + LayerNorm: one wave per row, lane owns 4 dims.
// ---------------------------------------------------------------------------
__global__ __launch_bounds__(256) void rga_ln_kernel(
    const float* __restrict__ ent, const float* __restrict__ proj,
    const float* __restrict__ gamma, const float* __restrict__ beta,
    float* __restrict__ out)
{
    const int lane = threadIdx.x & 31;
    const int wave = threadIdx.x >> 5;
    const int m = blockIdx.x * 8 + wave;

    const float* e = ent  + (size_t)m * 128;
    const float* p = proj + (size_t)m * 128;

    float x[4];
    float s = 0.0f;
    #pragma unroll
    for (int c = 0; c < 4; ++c) {
        x[c] = e[c * 32 + lane] + p[c * 32 + lane];
        s += x[c];
    }
    #pragma unroll
    for (int o = 16; o > 0; o >>= 1) s += __shfl_xor(s, o, 32);
    const float mu = s * (1.0f / 128.0f);

    float v = 0.0f;
    #pragma unroll
    for (int c = 0; c < 4; ++c) { const float d = x[c] - mu; v += d * d; }
    #pragma unroll
    for (int o = 16; o > 0; o >>= 1) v += __shfl_xor(v, o, 32);
    const float rstd = rsqrtf(v * (1.0f / 128.0f) + LN_EPS);

    float* orow = out + (size_t)m * 128;
    #pragma unroll
    for (int c = 0; c < 4; ++c) {
        const int col = c * 32 + lane;
        orow[col] = (x[c] - mu) * rstd * gamma[col] + beta[col];
    }
}

// ---------------------------------------------------------------------------
extern "C" void kernel_launch(void* const* d_in, const int* in_sizes, int n_in,
                              void* d_out, int out_size, void* d_ws, size_t ws_size,
                              hipStream_t stream)
{
    const float* entity   = (const float*)d_in[0];
    const int*   nbrIdx   = (const int*)  d_in[1];
    const int*   relIds   = (const int*)  d_in[2];
    const int*   nbrMask  = (const int*)  d_in[3];
    const float* Wq       = (const float*)d_in[4];
    const float* bq       = (const float*)d_in[5];
    const float* Wk       = (const float*)d_in[6];
    const float* bk       = (const float*)d_in[7];
    const float* Wv       = (const float*)d_in[8];
    const float* bv       = (const float*)d_in[9];
    const float* Wo       = (const float*)d_in[10];
    const float* bo       = (const float*)d_in[11];
    const float* relTable = (const float*)d_in[12];
    const float* gamma    = (const float*)d_in[13];
    const float* beta     = (const float*)d_in[14];
    float* out = (float*)d_out;

    // Workspace layout (floats): 5 x M*D + RelK  (~40.2 MB)
    const size_t MD = (size_t)M_NODES * D_DIM;     // 2,097,152
    float* ws   = (float*)d_ws;
    float* Qb   = ws;
    float* Kall = Qb   + MD;
    float* Vall = Kall + MD;
    float* Aout = Vall + MD;
    float* Tmp  = Aout + MD;
    float* RelK = Tmp  + MD;                        // 240*128 floats

    const dim3 blk(256);
    const int gemmGrid = (M_NODES / 16) / 8;        // 128 blocks, 8 waves each

    // Projections (gather factored out of the GEMMs):
    rga_gemm128_kernel<<<gemmGrid, blk, 0, stream>>>(entity, Wq, bq,      Qb,   M_NODES);
    rga_gemm128_kernel<<<gemmGrid, blk, 0, stream>>>(entity, Wk, nullptr, Kall, M_NODES); // bk folded into RelK
    rga_gemm128_kernel<<<gemmGrid, blk, 0, stream>>>(entity, Wv, bv,      Vall, M_NODES);
    rga_gemm128_kernel<<<2,        blk, 0, stream>>>(relTable, Wk, bk,    RelK, R_REL);

    // Gather-attention (one wave per node):
    rga_attn_kernel<<<M_NODES / 8, blk, 0, stream>>>(Qb, Kall, RelK, Vall,
                                                     nbrIdx, relIds, nbrMask, Aout);

    // Output projection + residual LayerNorm:
    rga_gemm128_kernel<<<gemmGrid, blk, 0, stream>>>(Aout, Wo, bo, Tmp, M_NODES);
    rga_ln_kernel<<<M_NODES / 8, blk, 0, stream>>>(entity, Tmp, gamma, beta, out);
}